// Router_3530463117598
// MI455X (gfx1250) — compile-verified
//
#include <hip/hip_runtime.h>
#include <math.h>
#include <stdint.h>

typedef __attribute__((ext_vector_type(2))) float v2f;
typedef __attribute__((ext_vector_type(8))) float v8f;

#define TOKENS   16384
#define HIDDEN   2048
#define EXPERTS  64
#define TOPK     8

#define WG_TOKENS 128      // tokens per workgroup (8 waves x 16 tokens)
#define KC        64       // K-chunk staged in LDS per iteration
#define WS        68       // padded LDS row stride (floats): 17*float4, 16B aligned, bank-skewed

#define XTILE   (16 * WS)            // floats per wave x-tile
#define XBUF    (8 * XTILE)          // 8704 floats: all 8 wave x-tiles
#define WBUF    (EXPERTS * WS)       // 4352 floats: weight chunk
#define STAGE   (XBUF + WBUF)        // 13056 floats per buffer
#define STAGE_B (STAGE * 4)          // bytes per buffer

// Async global -> LDS copy, 16B per lane (512B per wave32 instruction). ASYNCcnt-tracked.
__device__ __forceinline__ void async_cp16(uint32_t lds_off, const float* gptr) {
    asm volatile("global_load_async_to_lds_b128 %0, %1, off"
                 :: "v"(lds_off), "v"(gptr)
                 : "memory");
}

template <int N>
__device__ __forceinline__ void wait_async() {
#if defined(__has_builtin) && __has_builtin(__builtin_amdgcn_s_wait_asynccnt)
    __builtin_amdgcn_s_wait_asynccnt(N);
#else
    asm volatile("s_wait_asynccnt %0" :: "i"(N) : "memory");
#endif
}

// Fused router: fp32 WMMA gating GEMM + softmax + top-8, one pass over x,
// double-buffered LDS fed by async tensor-path copies.
__global__ __launch_bounds__(256) void router_fused_kernel(
    const float* __restrict__ x,       // [TOKENS, HIDDEN]
    const float* __restrict__ w,       // [EXPERTS, HIDDEN]
    float* __restrict__ out_w,         // [TOKENS, TOPK]
    int*   __restrict__ out_i)         // [TOKENS, TOPK]
{
    __shared__ __attribute__((aligned(16))) float sh[2 * STAGE];   // ~102 KB

    const int tid  = threadIdx.x;
    const int wave = tid >> 5;             // wave32
    const int lane = tid & 31;
    const int half = lane >> 4;            // selects K {0,1} vs {2,3} in WMMA A/B layout
    const int l16  = lane & 15;            // M (token) for A, N (expert) for B/C/D

    const int wg_tok = blockIdx.x * WG_TOKENS;
    const int wv_tok = wg_tok + wave * 16;

    // ---- precompute per-lane async-copy addresses (chunk k0 = 0, buffer 0) ----
    const uint32_t lds_base = (uint32_t)(uintptr_t)(void*)sh;   // low 32 bits = LDS offset

    const float* gx[8];  uint32_t lx[8];   // x tile: 8 async b128 per wave per chunk
    #pragma unroll
    for (int p = 0; p < 8; ++p) {
        const int idx = p * 32 + lane;
        const int r   = idx >> 4;          // token row 0..15
        const int q   = idx & 15;          // float4 column
        gx[p] = x + (size_t)(wv_tok + r) * HIDDEN + q * 4;
        lx[p] = lds_base + (uint32_t)(wave * XTILE + r * WS + q * 4) * 4u;
    }
    const float* gw[4];  uint32_t lw[4];   // weight chunk: wave owns experts [wave*8, wave*8+8)
    #pragma unroll
    for (int p = 0; p < 4; ++p) {
        const int idx = p * 32 + lane;
        const int r   = wave * 8 + (idx >> 4);
        const int q   = idx & 15;
        gw[p] = w + (size_t)r * HIDDEN + q * 4;
        lw[p] = lds_base + (uint32_t)(XBUF + r * WS + q * 4) * 4u;
    }

    // issue all 12 async copies for chunk at byte offset k0*4 into buffer `buf`
    auto issue_chunk = [&](int buf, int k0) {
        const uint32_t boff = (uint32_t)buf * STAGE_B;
        #pragma unroll
        for (int p = 0; p < 8; ++p) async_cp16(lx[p] + boff, gx[p] + k0);
        #pragma unroll
        for (int p = 0; p < 4; ++p) async_cp16(lw[p] + boff, gw[p] + k0);
    };

    v8f acc[4];
    #pragma unroll
    for (int et = 0; et < 4; ++et)
        #pragma unroll
        for (int j = 0; j < 8; ++j) acc[et][j] = 0.0f;

    issue_chunk(0, 0);                      // prologue: fill buffer 0
    int cur = 0;

    for (int k0 = 0; k0 < HIDDEN; k0 += KC) {
        const int nk0 = k0 + KC;
        if (nk0 < HIDDEN) {
            issue_chunk(cur ^ 1, nk0);      // prefetch next chunk into the other buffer
            wait_async<12>();               // our current-chunk copies have landed
        } else {
            wait_async<0>();
        }
        __syncthreads();                    // all waves' current-chunk copies visible

        const float* xs  = sh + cur * STAGE + wave * XTILE;   // [16][WS]
        const float* wsm = sh + cur * STAGE + XBUF;           // [64][WS]

        #pragma unroll
        for (int k4 = 0; k4 < KC / 4; ++k4) {
            const int kk = k4 * 4 + half * 2;
            const v2f a = *(const v2f*)(xs + l16 * WS + kk);          // A: x[tok=l16][k..k+1]
            #pragma unroll
            for (int et = 0; et < 4; ++et) {
                const v2f b = *(const v2f*)(wsm + (et * 16 + l16) * WS + kk);
                acc[et] = __builtin_amdgcn_wmma_f32_16x16x4_f32(
                    false, a, false, b, (short)0, acc[et], false, false);
            }
        }
        __syncthreads();                    // readers done: buffer reusable next iteration
        cur ^= 1;
    }

    // ---- scatter logits to LDS: lg[token_in_wg][expert], stride 65 (bank-skewed) ----
    float* lg = sh;                          // 128*65 = 8320 floats, fits in buffer 0
    #pragma unroll
    for (int et = 0; et < 4; ++et)
        #pragma unroll
        for (int j = 0; j < 8; ++j) {
            const int tokw = wave * 16 + j + 8 * half;   // C/D layout: M = j + 8*half
            const int e    = et * 16 + l16;              // N = l16
            lg[tokw * 65 + e] = acc[et][j];
        }
    __syncthreads();

    // ---- softmax + top-8 per token (threads 0..127, one token each) ----
    if (tid < WG_TOKENS) {
        const int tok = wg_tok + tid;
        const float* row = lg + tid * 65;

        float mx = row[0];
        #pragma unroll
        for (int e = 1; e < EXPERTS; ++e) mx = fmaxf(mx, row[e]);
        float sum = 0.0f;
        #pragma unroll
        for (int e = 0; e < EXPERTS; ++e) sum += expf(row[e] - mx);
        const float inv = 1.0f / sum;

        unsigned long long taken = 0ull;
        #pragma unroll
        for (int i = 0; i < TOPK; ++i) {
            float best = -INFINITY;
            int   bi   = 0;
            for (int e = 0; e < EXPERTS; ++e) {
                // strict '>' keeps lowest index on ties, matching lax.top_k
                if (!((taken >> e) & 1ull) && row[e] > best) { best = row[e]; bi = e; }
            }
            taken |= (1ull << bi);
            out_w[(size_t)tok * TOPK + i] = expf(best - mx) * inv;
            out_i[(size_t)tok * TOPK + i] = bi;
        }
    }
}

extern "C" void kernel_launch(void* const* d_in, const int* in_sizes, int n_in,
                              void* d_out, int out_size, void* d_ws, size_t ws_size,
                              hipStream_t stream) {
    const float* x = (const float*)d_in[0];   // [16384, 2048] fp32
    const float* w = (const float*)d_in[1];   // [64, 2048] fp32

    float* out_w = (float*)d_out;                                   // [16384, 8] fp32
    int*   out_i = (int*)((float*)d_out + (size_t)TOKENS * TOPK);   // [16384, 8] int32

    dim3 grid(TOKENS / WG_TOKENS);   // 128 workgroups
    dim3 block(256);                 // 8 wave32s
    router_fused_kernel<<<grid, block, 0, stream>>>(x, w, out_w, out_i);
}